// PCCLoss_42460046688484
// MI455X (gfx1250) — compile-verified
//
#include <hip/hip_runtime.h>

typedef __attribute__((ext_vector_type(2))) float v2f;
typedef __attribute__((ext_vector_type(8))) float v8f;

#define SAMPLES 64
#define ELEMS   196608            // 3*256*256 per sample
#define VEC4S   (ELEMS / 4)       // 49152 float4 per sample
#define CHUNKS  16                // blocks per sample
#define THREADS 512               // 16 waves (wave32)
#define WAVES   (THREADS / 32)
#define VEC4S_PER_CHUNK (VEC4S / CHUNKS)   // 3072
#define ITERS   (VEC4S_PER_CHUNK / THREADS) // 6

// Kernel 1: per-(sample,chunk) partial sums of {a, b, a*a, b*b, a*b}.
// Streaming b128 loads, fp32 accumulation. Cross-wave reduction uses
// V_WMMA_F32_16X16X4_F32 with B = ones (row-sum engine): A rows = the 5 sums
// (rows 5..15 zero-padded in LDS), A cols = 4 waves per accumulating WMMA,
// 4 WMMAs cover all 16 waves. No exec-predicated loads: the LDS tile is
// fully initialized so every lane loads unconditionally.
__global__ __launch_bounds__(THREADS)
void pcc_partial_kernel(const float* __restrict__ ya,
                        const float* __restrict__ yb,
                        float* __restrict__ part) {
  const int blk    = blockIdx.x;
  const int sample = blk >> 4;          // blk / CHUNKS
  const int chunk  = blk & (CHUNKS - 1);

  const float4* __restrict__ a4 =
      reinterpret_cast<const float4*>(ya) + (size_t)sample * VEC4S + (size_t)chunk * VEC4S_PER_CHUNK;
  const float4* __restrict__ b4 =
      reinterpret_cast<const float4*>(yb) + (size_t)sample * VEC4S + (size_t)chunk * VEC4S_PER_CHUNK;

  float sa = 0.f, sb = 0.f, saa = 0.f, sbb = 0.f, sab = 0.f;

#pragma unroll
  for (int it = 0; it < ITERS; ++it) {
    const int idx = it * THREADS + threadIdx.x;   // coalesced b128 per wave
    const float4 av = a4[idx];
    const float4 bv = b4[idx];
    sa += (av.x + av.y) + (av.z + av.w);
    sb += (bv.x + bv.y) + (bv.z + bv.w);
    saa = fmaf(av.x, av.x, saa); saa = fmaf(av.y, av.y, saa);
    saa = fmaf(av.z, av.z, saa); saa = fmaf(av.w, av.w, saa);
    sbb = fmaf(bv.x, bv.x, sbb); sbb = fmaf(bv.y, bv.y, sbb);
    sbb = fmaf(bv.z, bv.z, sbb); sbb = fmaf(bv.w, bv.w, sbb);
    sab = fmaf(av.x, bv.x, sab); sab = fmaf(av.y, bv.y, sab);
    sab = fmaf(av.z, bv.z, sab); sab = fmaf(av.w, bv.w, sab);
  }

  // Intra-wave xor-butterfly: afterwards EVERY lane holds the wave totals.
#pragma unroll
  for (int off = 16; off > 0; off >>= 1) {
    sa  += __shfl_xor(sa,  off, 32);
    sb  += __shfl_xor(sb,  off, 32);
    saa += __shfl_xor(saa, off, 32);
    sbb += __shfl_xor(sbb, off, 32);
    sab += __shfl_xor(sab, off, 32);
  }

  // Zero-padded staging tile: wsum[wave][row], rows 5..15 = 0.
  __shared__ float wsum[WAVES][16];
  const int wave = threadIdx.x >> 5;
  const int lane = threadIdx.x & 31;
  if (lane < 16) {
    float v = 0.0f;
    if      (lane == 0) v = sa;
    else if (lane == 1) v = sb;
    else if (lane == 2) v = saa;
    else if (lane == 3) v = sbb;
    else if (lane == 4) v = sab;
    wsum[wave][lane] = v;     // lanes 5..15 write the zero padding
  }
  __syncthreads();

  // Wave 0 only: EXEC is all-1s for its 32 lanes (WMMA requirement).
  if (threadIdx.x < 32) {
    // A (16x4 f32) per-lane layout: lanes 0-15 -> M=lane, K={0,1};
    //                               lanes 16-31 -> M=lane-16, K={2,3}.
    const int row = lane & 15;           // M = sum index
    const int kb  = (lane >> 4) << 1;    // K base: 0 or 2
    v2f bones; bones.x = 1.0f; bones.y = 1.0f;   // B = ones -> D cols = row sums
    v8f c = {};
#pragma unroll
    for (int j = 0; j < 4; ++j) {        // waves 4j .. 4j+3 per WMMA
      const int w0 = j * 4 + kb;
      v2f aV;
      aV.x = wsum[w0 + 0][row];          // unconditional ds_load_b32
      aV.y = wsum[w0 + 1][row];
      c = __builtin_amdgcn_wmma_f32_16x16x4_f32(
          /*neg_a=*/false, aV, /*neg_b=*/false, bones,
          /*c_mod=*/(short)0, c, /*reuse_a=*/false, /*reuse_b=*/false);
    }
    // C/D layout: lane 0 holds (M=0..7, N=0) in c[0..7] -> c[s] = total sum s.
    if (lane == 0) {
      float* o = part + (size_t)blk * 5;
      o[0] = c[0]; o[1] = c[1]; o[2] = c[2]; o[3] = c[3]; o[4] = c[4];
    }
  }
}

// Kernel 2: fold chunk partials per sample, compute pcc, average, 1 - mean.
__global__ __launch_bounds__(64)
void pcc_finalize_kernel(const float* __restrict__ part, float* __restrict__ out) {
  __shared__ float wred[2];
  const int i = threadIdx.x;   // sample index, 0..63

  float Sa = 0.f, Sb = 0.f, Saa = 0.f, Sbb = 0.f, Sab = 0.f;
  for (int c = 0; c < CHUNKS; ++c) {
    const float* p = part + ((size_t)i * CHUNKS + c) * 5;
    Sa += p[0]; Sb += p[1]; Saa += p[2]; Sbb += p[3]; Sab += p[4];
  }

  const float invN = 1.0f / (float)ELEMS;
  const float cov = Sab - Sa * Sb * invN;
  const float va  = Saa - Sa * Sa * invN;
  const float vb  = Sbb - Sb * Sb * invN;
  float pcc = cov * rsqrtf(va * vb);

#pragma unroll
  for (int off = 16; off > 0; off >>= 1)
    pcc += __shfl_xor(pcc, off, 32);

  const int lane = i & 31, wave = i >> 5;
  if (lane == 0) wred[wave] = pcc;
  __syncthreads();
  if (i == 0) out[0] = 1.0f - (wred[0] + wred[1]) * (1.0f / (float)SAMPLES);
}

extern "C" void kernel_launch(void* const* d_in, const int* in_sizes, int n_in,
                              void* d_out, int out_size, void* d_ws, size_t ws_size,
                              hipStream_t stream) {
  const float* ya = (const float*)d_in[0];
  const float* yb = (const float*)d_in[1];
  float* part = (float*)d_ws;                 // 1024 * 5 floats = 20 KB
  float* out  = (float*)d_out;

  pcc_partial_kernel<<<SAMPLES * CHUNKS, THREADS, 0, stream>>>(ya, yb, part);
  pcc_finalize_kernel<<<1, 64, 0, stream>>>(part, out);
}